// Upsample_51573967290886
// MI455X (gfx1250) — compile-verified
//
#include <hip/hip_runtime.h>

// Upsample2d (up=2, 4x4 FIR, gain=4) as polyphase 2x2-quad kernel.
// x: [16,128,128,128] f32 (NCHW), kernel: [4,4] f32, out: [16,128,256,256] f32.
// HBM-bound (~0.67 GB traffic -> ~29us floor @ 23.3 TB/s).
// gfx1250 paths: async global->LDS staging (b128 wide path for interior tiles,
// b32 predicated path at boundaries), s_wait_asynccnt, NT float4 output stores.

#define TILE   32
#define HALO   34        // TILE + 2 (one-element halo each side)
#define PITCH  44        // multiple of 4 floats -> 16B-aligned LDS rows for b128
#define WIDE   40        // b128 path stages 40 cols (tj0-4 .. tj0+35)
#define H_IN   128
#define W_IN   128
#define H_OUT  256
#define W_OUT  256
#define NTHREADS 256

typedef float f4 __attribute__((ext_vector_type(4)));

__global__ __launch_bounds__(NTHREADS) void upsample2d_poly_kernel(
    const float* __restrict__ x,
    const float* __restrict__ kern,
    float* __restrict__ out)
{
    __shared__ float tile[HALO * PITCH];

    const int tid = threadIdx.x;
    const int p   = blockIdx.z;                 // plane = n*C + c
    const int ti0 = blockIdx.y * TILE;          // input tile origin (rows)
    const int tj0 = blockIdx.x * TILE;          // input tile origin (cols)

    // Per-phase weights: W[ry][rx][a][b] = 4 * K[3-2a-ry][3-2b-rx]
    // (uniform -> scalar loads, issued early so latency is hidden)
    float w[2][2][2][2];
#pragma unroll
    for (int ry = 0; ry < 2; ++ry)
#pragma unroll
        for (int rx = 0; rx < 2; ++rx)
#pragma unroll
            for (int a = 0; a < 2; ++a)
#pragma unroll
                for (int b = 0; b < 2; ++b)
                    w[ry][rx][a][b] =
                        4.0f * kern[(3 - 2 * a - ry) * 4 + (3 - 2 * b - rx)];

    // Wide (b128) staging path is legal only when all staged columns are
    // in-plane: tj0-4 >= 0 and tj0+35 < 128.
    const bool colInterior = (tj0 != 0) && (tj0 + TILE < W_IN);
    const bool rowBoundary = (ti0 == 0) || (ti0 + TILE >= H_IN);
    const bool needZero    = (!colInterior) || rowBoundary;
    // LDS column holding global column (tj0 + qx - 1) for quad column qx:
    const int  cofs        = colInterior ? 3 : 0;

    if (needZero) {
        for (int k2 = tid; k2 < HALO * PITCH; k2 += NTHREADS)
            tile[k2] = 0.0f;
    }
    __syncthreads();   // ds stores complete before async writes can land

    const float* xp = x + (size_t)p * (H_IN * W_IN);

#if defined(__HIP_DEVICE_COMPILE__)
    __attribute__((address_space(3))) float* lds =
        (__attribute__((address_space(3))) float*)&tile[0];
#endif

    if (colInterior) {
        // 34 rows x 10 b128 chunks: global cols tj0-4 .. tj0+35 -> LDS cols 0..39.
        for (int k2 = tid; k2 < HALO * (WIDE / 4); k2 += NTHREADS) {
            const int r  = k2 / (WIDE / 4);
            const int ch = k2 - r * (WIDE / 4);
            const int gi = ti0 - 1 + r;
            if ((unsigned)gi < (unsigned)H_IN) {
#if defined(__HIP_DEVICE_COMPILE__)
                const unsigned loff =
                    (unsigned)(unsigned long long)(lds + (r * PITCH + ch * 4));
                const float* ga = xp + (gi * W_IN + (tj0 - 4) + ch * 4);
                asm volatile("global_load_async_to_lds_b128 %0, %1, off"
                             :
                             : "v"(loff), "v"(ga)
                             : "memory");
#else
                for (int e = 0; e < 4; ++e)
                    tile[r * PITCH + ch * 4 + e] =
                        xp[gi * W_IN + (tj0 - 4) + ch * 4 + e];
#endif
            }
        }
    } else {
        // Boundary tiles: per-element b32 path, predicated; halo stays zero.
        for (int k2 = tid; k2 < HALO * HALO; k2 += NTHREADS) {
            const int r  = k2 / HALO;
            const int c  = k2 - r * HALO;
            const int gi = ti0 - 1 + r;
            const int gj = tj0 - 1 + c;
            if (((unsigned)gi < (unsigned)H_IN) & ((unsigned)gj < (unsigned)W_IN)) {
#if defined(__HIP_DEVICE_COMPILE__)
                const unsigned loff =
                    (unsigned)(unsigned long long)(lds + (r * PITCH + c));
                const float* ga = xp + (gi * W_IN + gj);
                asm volatile("global_load_async_to_lds_b32 %0, %1, off"
                             :
                             : "v"(loff), "v"(ga)
                             : "memory");
#else
                tile[r * PITCH + c] = xp[gi * W_IN + gj];
#endif
            }
        }
    }
#if defined(__HIP_DEVICE_COMPILE__)
    asm volatile("s_wait_asynccnt 0" ::: "memory");
#endif
    __syncthreads();

    // Compute: 32x32 quads. Thread = (tyb 0..15, txq even 0..30); walks 2
    // quad-rows, 2 adjacent quad-cols -> one float4 NT store per output row.
    const int txq = (tid & 15) << 1;   // first quad column (even)
    const int tyb = tid >> 4;          // 0..15
    float* op = out + (size_t)p * (H_OUT * W_OUT);

    auto ph = [&](int ry, int rx, float s00, float s01, float s10, float s11) {
        return w[ry][rx][0][0] * s00 + w[ry][rx][0][1] * s01 +
               w[ry][rx][1][0] * s10 + w[ry][rx][1][1] * s11;
    };

    float A0[4], A1[4], A2[4];
    {
        const int b0 = (tyb * 2) * PITCH + txq + cofs;
#pragma unroll
        for (int e = 0; e < 4; ++e) {
            A0[e] = tile[b0 + e];
            A1[e] = tile[b0 + PITCH + e];
        }
    }
#pragma unroll
    for (int it = 0; it < 2; ++it) {
        const int qy = tyb * 2 + it;
        const int b2 = (qy + 2) * PITCH + txq + cofs;
#pragma unroll
        for (int e = 0; e < 4; ++e) A2[e] = tile[b2 + e];

        f4 v0, v1;
        // output row oy: phases (0,0),(0,1) of quad q0 then q1
        v0.x = ph(0, 0, A0[0], A0[1], A1[0], A1[1]);
        v0.y = ph(0, 1, A0[1], A0[2], A1[1], A1[2]);
        v0.z = ph(0, 0, A0[1], A0[2], A1[1], A1[2]);
        v0.w = ph(0, 1, A0[2], A0[3], A1[2], A1[3]);
        // output row oy+1: phases (1,0),(1,1)
        v1.x = ph(1, 0, A1[0], A1[1], A2[0], A2[1]);
        v1.y = ph(1, 1, A1[1], A1[2], A2[1], A2[2]);
        v1.z = ph(1, 0, A1[1], A1[2], A2[1], A2[2]);
        v1.w = ph(1, 1, A1[2], A1[3], A2[2], A2[3]);

        const int oy = 2 * (ti0 + qy);
        const int ox = 2 * (tj0 + txq);           // multiple of 4 -> 16B aligned
        __builtin_nontemporal_store(v0, (f4*)(op + (size_t)oy * W_OUT + ox));
        __builtin_nontemporal_store(v1, (f4*)(op + (size_t)(oy + 1) * W_OUT + ox));

#pragma unroll
        for (int e = 0; e < 4; ++e) { A0[e] = A1[e]; A1[e] = A2[e]; }
    }
}

extern "C" void kernel_launch(void* const* d_in, const int* in_sizes, int n_in,
                              void* d_out, int out_size, void* d_ws, size_t ws_size,
                              hipStream_t stream) {
    const float* x    = (const float*)d_in[0];
    const float* kern = (const float*)d_in[1];
    float* out        = (float*)d_out;

    const int planes = in_sizes[0] / (H_IN * W_IN);   // N*C = 2048
    dim3 grid(W_IN / TILE, H_IN / TILE, planes);      // (4, 4, 2048)
    dim3 block(NTHREADS);
    upsample2d_poly_kernel<<<grid, block, 0, stream>>>(x, kern, out);
}